// ConvCaps_76510547411540
// MI455X (gfx1250) — compile-verified
//
#include <hip/hip_runtime.h>
#include <hip/hip_bf16.h>
#include <math.h>

// ---------------------------------------------------------------------------
// ConvCaps EM-routing for MI455X (gfx1250, wave32).
// Phase 1: votes via V_WMMA_F32_16X16X4_F32 (M=128,K=4,N=144 GEMM per (b,BKL)),
//          stored bf16 so the full V tensor (170 MB) is L2-resident (192 MB L2).
//          Internal pose-matrix index is h' = j*4+i so each lane's 4 D-regs
//          pack into one b64 store (2 stores/tile instead of 8).
// Phase 2: 3 EM iterations, fully fused: R is never materialized; it is
//          recomputed on the fly from previous-iteration (mu, 1/sigma, logterm)
//          plus a per-row apsum pass.  6 total streaming passes over V.
//          k_mstep stages each 9 KB V column into LDS via async-to-LDS b128
//          (ASYNCcnt) when available; k_apsum uses explicit b128 loads.
// ---------------------------------------------------------------------------

#define B_N      16
#define WIN      14
#define WOUT     6
#define BCAPS    32
#define CCAPS    32
#define BKK      288      // BCAPS*3*3
#define CWW      1152     // CCAPS*WOUT*WOUT
#define WW       36       // WOUT*WOUT
#define HH       16
#define EPSV     1e-10f
#define LOG2PI   1.8378770664093453f

#if __has_builtin(__builtin_amdgcn_global_load_async_to_lds_b128)
#define USE_ASYNC_LDS 1
#else
#define USE_ASYNC_LDS 0
#endif

typedef __attribute__((ext_vector_type(2))) float v2f;
typedef __attribute__((ext_vector_type(8))) float v8f;
typedef __attribute__((ext_vector_type(4))) int   v4i;

static __device__ __forceinline__ float bf2f(unsigned short u) {
    union { unsigned int i; float f; } c;
    c.i = ((unsigned int)u) << 16;
    return c.f;
}

// ---------------------------------------------------------------------------
// Phase 1: votes.  One block per (bkl, b).  8 waves, wave = M-tile (M=128).
// A (16x4 f32)  = W[bkl][C][i][k]   rows m=(C,i), cols k
// B (4x16 f32)  = pose window       rows k, cols n=(wv,j)
// D (16x16 f32) -> V[b][cww][bkl][h'],  h' = j*4+i  (internal order)
// A layout (ISA 7.12.2): lanes 0-15 M=0-15 K={0,1}, lanes 16-31 M=0-15 K={2,3}
// ---------------------------------------------------------------------------
__global__ __launch_bounds__(256) void k_votes(
    const float* __restrict__ poses,   // (16, 512, 14, 14)
    const float* __restrict__ W,       // (32,3,3,32,4,4)
    __hip_bfloat16* __restrict__ V)    // (16, 1152, 288, 16)
{
    const int bkl = blockIdx.x;            // 0..287  (Bc*9 + Kr*3 + Lc)
    const int b   = blockIdx.y;            // 0..15
    const int Bc  = bkl / 9;
    const int Kr  = (bkl / 3) % 3;
    const int Lc  = bkl % 3;

    const int lane = threadIdx.x & 31;
    const int wave = threadIdx.x >> 5;     // M tile 0..7
    const int mrow = lane & 15;
    const int kb   = (lane >> 4) << 1;     // K base: 0 (lo lanes) or 2 (hi lanes)

    // ---- A operand: W row m = wave*16 + mrow, elems k = kb, kb+1 (contiguous)
    const int m = wave * 16 + mrow;                    // m = C*4 + i
    const float* wp = W + (size_t)bkl * 512 + (size_t)m * 4 + kb;
    v2f A;
    A.x = wp[0];
    A.y = wp[1];

    const int mbase = wave * 16 + ((lane >> 4) << 3);  // +8 for hi half lanes
    const int Cc0   = mbase >> 2;                      // C of D-regs r=0..3

    for (int tn = 0; tn < 9; ++tn) {                   // N = 144 -> 9 tiles
        // ---- B operand: B[k][n], n = tn*16 + mrow, per-lane rows kb, kb+1
        const int n  = tn * 16 + mrow;                 // n = wv*4 + j  (<144)
        const int wv = n >> 2;
        const int j  = n & 3;
        const int y  = (wv / WOUT) * 2 + Kr;
        const int x  = (wv % WOUT) * 2 + Lc;
        // poses[b][(k*4+j)*32 + Bc][y][x]; channel stride = 32*196 = 6272
        const size_t pbase = (((size_t)b * HH) * BCAPS + Bc) * 196 + (size_t)y * WIN + x;
        v2f Bv;
        Bv.x = poses[pbase + (size_t)((kb + 0) * 4 + j) * 6272];
        Bv.y = poses[pbase + (size_t)((kb + 1) * 4 + j) * 6272];

        v8f acc = {};
        acc = __builtin_amdgcn_wmma_f32_16x16x4_f32(
            /*neg_a=*/false, A, /*neg_b=*/false, Bv,
            /*c_mod=*/(short)0, acc, /*reuse_a=*/false, /*reuse_b=*/false);

        // ---- store D: reg r holds M = mbase + r -> (C = Cc0 + (r>>2), i = r&3)
        // internal h' = j*4 + i, so r=0..3 are 4 consecutive bf16 -> one b64.
        unsigned short pk[8];
        #pragma unroll
        for (int r = 0; r < 8; ++r) {
            __hip_bfloat16 t = __float2bfloat16(acc[r]);
            __builtin_memcpy(&pk[r], &t, 2);
        }
        const size_t base0 =
            (((size_t)b * CWW + (size_t)Cc0 * WW + wv) * BKK + bkl) * HH + j * 4;
        const size_t base1 = base0 + (size_t)WW * BKK * HH;   // C = Cc0 + 1
        uint2 w0, w1;
        __builtin_memcpy(&w0, pk, 8);
        __builtin_memcpy(&w1, pk + 4, 8);
        *reinterpret_cast<uint2*>(V + base0) = w0;
        *reinterpret_cast<uint2*>(V + base1) = w1;
    }
}

// ---------------------------------------------------------------------------
// Gather windowed activations: aw[b][bkk][wv] (shared across C).
// ---------------------------------------------------------------------------
__global__ void k_aw(const float* __restrict__ act,   // (16,32,14,14)
                     float* __restrict__ aw)          // (16,288,36)
{
    const int idx = blockIdx.x * blockDim.x + threadIdx.x;
    if (idx >= B_N * BKK * WW) return;
    const int wv  = idx % WW;
    const int bkk = (idx / WW) % BKK;
    const int b   = idx / (WW * BKK);
    const int Bc  = bkk / 9;
    const int Kr  = (bkk / 3) % 3;
    const int Lc  = bkk % 3;
    const int y   = (wv / WOUT) * 2 + Kr;
    const int x   = (wv % WOUT) * 2 + Lc;
    aw[idx] = act[((size_t)b * BCAPS + Bc) * 196 + (size_t)y * WIN + x];
}

// ---------------------------------------------------------------------------
// apsum[b][bkk] = sum_cww a_old[cww] * p(bkk,cww)   (E-step row sums)
// p recomputed from previous-iteration stats. One block per (bkk, b).
// Explicit b128 loads of the 32-byte (bkk-row, cww) granule.
// ---------------------------------------------------------------------------
__global__ __launch_bounds__(256) void k_apsum(
    const __hip_bfloat16* __restrict__ V,
    const float* __restrict__ mu_o, const float* __restrict__ is_o,
    const float* __restrict__ c_o,  const float* __restrict__ a_old,
    float* __restrict__ apsum)
{
    const int bkk = blockIdx.x;
    const int b   = blockIdx.y;
    const int tid = threadIdx.x;

    const __hip_bfloat16* vb = V + ((size_t)b * CWW * BKK + bkk) * HH;
    float acc = 0.f;
    for (int cww = tid; cww < CWW; cww += 256) {
        const uint4* vp =
            reinterpret_cast<const uint4*>(vb + (size_t)cww * (BKK * HH));
        __builtin_prefetch((const void*)(vp + (size_t)256 * BKK), 0, 0);
        const uint4 q0 = vp[0];
        const uint4 q1 = vp[1];
        unsigned int qw[8] = {q0.x, q0.y, q0.z, q0.w, q1.x, q1.y, q1.z, q1.w};
        const size_t sb = ((size_t)b * CWW + cww) * HH;
        float p = 0.f;
        #pragma unroll
        for (int k = 0; k < 8; ++k) {
            const float v0 = bf2f((unsigned short)(qw[k] & 0xffffu));
            const float v1 = bf2f((unsigned short)(qw[k] >> 16));
            const int h = 2 * k;
            float z = (v0 + EPSV - mu_o[sb + h]) * is_o[sb + h];
            p += __expf(-0.5f * z * z + c_o[sb + h]);
            z = (v1 + EPSV - mu_o[sb + h + 1]) * is_o[sb + h + 1];
            p += __expf(-0.5f * z * z + c_o[sb + h + 1]);
        }
        acc += a_old[(size_t)b * CWW + cww] * p;
    }
    __shared__ float red[256];
    red[tid] = acc;
    __syncthreads();
    for (int s = 128; s > 0; s >>= 1) {
        if (tid < s) red[tid] += red[tid + s];
        __syncthreads();
    }
    if (tid == 0) apsum[b * BKK + bkk] = red[0];
}

// ---------------------------------------------------------------------------
// M-step (+cost/activation), one block per (cww, b).  Threads: h = tid&15,
// group g = tid>>4 strides over bkk.  The contiguous 9216-B V column is
// staged into LDS via async-to-LDS b128 (ASYNCcnt) when available.
// iter==0: R uniform.  iter>0: R = a_old*p/apsum + eps recomputed on the fly.
// iter==2: writes final outputs (h' = j*4+i un-permuted to i*4+j).
// ---------------------------------------------------------------------------
__global__ __launch_bounds__(256) void k_mstep(
    const __hip_bfloat16* __restrict__ V,
    const float* __restrict__ aw,     const float* __restrict__ apsum,
    const float* __restrict__ mu_o,   const float* __restrict__ is_o,
    const float* __restrict__ c_o,    const float* __restrict__ a_old,
    float* __restrict__ mu_n, float* __restrict__ is_n,
    float* __restrict__ c_n,  float* __restrict__ a_n,
    const float* __restrict__ beta_v, const float* __restrict__ beta_a,
    const float* __restrict__ lambda_,
    float* __restrict__ out_poses, float* __restrict__ out_act, int iter)
{
    const int cww = blockIdx.x;
    const int b   = blockIdx.y;
    const int tid = threadIdx.x;
    const int h   = tid & 15;
    const int g   = tid >> 4;          // 16 groups of 16 lanes
    const int wv  = cww % WW;

    const size_t sb = ((size_t)b * CWW + cww) * HH + h;

    float m_o = 0.f, i_o = 0.f, cc_o = 0.f, aold = 0.f;
    if (iter > 0) {
        m_o  = mu_o[sb];
        i_o  = is_o[sb];
        cc_o = c_o[sb];
        aold = a_old[(size_t)b * CWW + cww];
    }

    // ---- stage the V column (BKK*HH bf16 = 9216 B, contiguous) into LDS ----
    __shared__ __align__(16) unsigned short svv[BKK * HH];
    const char* vbc = reinterpret_cast<const char*>(
        V + ((size_t)b * CWW + cww) * (BKK * HH));
#if USE_ASYNC_LDS
    for (int off = tid * 16; off < BKK * HH * 2; off += 256 * 16) {
        __builtin_amdgcn_global_load_async_to_lds_b128(
            (__attribute__((address_space(1))) v4i*)(vbc + off),
            (__attribute__((address_space(3))) v4i*)((char*)svv + off),
            0, 0);
    }
#if __has_builtin(__builtin_amdgcn_s_wait_asynccnt)
    __builtin_amdgcn_s_wait_asynccnt(0);
#else
    asm volatile("s_wait_asynccnt 0x0" ::: "memory");
#endif
    __syncthreads();
#else
    {
        const uint4* src = reinterpret_cast<const uint4*>(vbc);
        uint4* dst = reinterpret_cast<uint4*>(svv);
        for (int t = tid; t < (BKK * HH * 2) / 16; t += 256) dst[t] = src[t];
        __syncthreads();
    }
#endif

    float s0 = 0.f, s1 = 0.f, s2 = 0.f;
    for (int bkk = g; bkk < BKK; bkk += 16) {          // 18 uniform steps
        const float v = bf2f(svv[bkk * HH + h]);
        float R;
        if (iter == 0) {
            R = 1.0f / (float)CWW;
        } else {
            const float z = (v + EPSV - m_o) * i_o;
            float p = __expf(-0.5f * z * z + cc_o);
            p += __shfl_xor(p, 1, 16);
            p += __shfl_xor(p, 2, 16);
            p += __shfl_xor(p, 4, 16);
            p += __shfl_xor(p, 8, 16);
            R = aold * p / apsum[b * BKK + bkk] + EPSV;
        }
        const float rw = R * aw[((size_t)b * BKK + bkk) * WW + wv];
        if (h == 0) s0 += rw;
        s1 += rw * v;
        s2 += rw * v * v;
    }

    // Cross-group reduction (same h across the 16 groups)
    __shared__ float red[256];
    red[tid] = s1; __syncthreads();
    if (g == 0) { float t = 0.f; for (int gg = 0; gg < 16; ++gg) t += red[gg * 16 + h]; s1 = t; }
    __syncthreads();
    red[tid] = s2; __syncthreads();
    if (g == 0) { float t = 0.f; for (int gg = 0; gg < 16; ++gg) t += red[gg * 16 + h]; s2 = t; }
    __syncthreads();
    red[tid] = s0; __syncthreads();
    if (tid == 0) { float t = 0.f; for (int gg = 0; gg < 16; ++gg) t += red[gg * 16]; red[0] = t; }
    __syncthreads();
    s0 = red[0];

    if (g == 0) {
        const float mu   = s1 / s0;
        const float sig2 = fmaxf(s2 / s0 - mu * mu, 1e-20f);
        const float sig  = sqrtf(sig2);
        const float invs = 1.0f / sig;
        const float cc   = -__logf(sig) - 0.5f * LOG2PI;
        mu_n[sb] = mu;
        is_n[sb] = invs;
        c_n[sb]  = cc;

        const int Cc = cww / WW;
        float costh = (beta_v[Cc] + __logf(sig + EPSV)) * s0;
        costh += __shfl_xor(costh, 1, 16);
        costh += __shfl_xor(costh, 2, 16);
        costh += __shfl_xor(costh, 4, 16);
        costh += __shfl_xor(costh, 8, 16);
        if (h == 0) {
            const float aval =
                1.0f / (1.0f + __expf(-lambda_[0] * (beta_a[Cc] - costh)));
            a_n[(size_t)b * CWW + cww] = aval;
            if (iter == 2) out_act[(size_t)b * CWW + cww] = aval;
        }
        if (iter == 2) {
            // internal h' = j*4+i  ->  reference h = i*4+j
            const int href = (h & 3) * 4 + (h >> 2);
            out_poses[((size_t)b * CWW + cww) * HH + href] = mu;
        }
    }
}

// ---------------------------------------------------------------------------
extern "C" void kernel_launch(void* const* d_in, const int* in_sizes, int n_in,
                              void* d_out, int out_size, void* d_ws, size_t ws_size,
                              hipStream_t stream)
{
    const float* poses   = (const float*)d_in[0];
    const float* act     = (const float*)d_in[1];
    const float* W       = (const float*)d_in[2];
    const float* beta_v  = (const float*)d_in[3];
    const float* beta_a  = (const float*)d_in[4];
    const float* lambda_ = (const float*)d_in[5];

    float* out_poses = (float*)d_out;                       // 16*1152*16
    float* out_act   = out_poses + (size_t)B_N * CWW * HH;  // 16*1152

    // ---- workspace carve-up
    char* ws = (char*)d_ws;
    size_t off = 0;
    __hip_bfloat16* V = (__hip_bfloat16*)(ws + off);
    off += (size_t)B_N * CWW * BKK * HH * sizeof(__hip_bfloat16);   // 170 MB
    float* aw = (float*)(ws + off);     off += (size_t)B_N * BKK * WW * sizeof(float);
    float* apsum = (float*)(ws + off);  off += (size_t)B_N * BKK * sizeof(float);
    float* muB[2]; float* isB[2]; float* cB[2]; float* aB[2];
    for (int i = 0; i < 2; ++i) {
        muB[i] = (float*)(ws + off); off += (size_t)B_N * CWW * HH * sizeof(float);
        isB[i] = (float*)(ws + off); off += (size_t)B_N * CWW * HH * sizeof(float);
        cB[i]  = (float*)(ws + off); off += (size_t)B_N * CWW * HH * sizeof(float);
        aB[i]  = (float*)(ws + off); off += (size_t)B_N * CWW * sizeof(float);
    }

    // Phase 1: votes (WMMA) + activation windows
    k_votes<<<dim3(BKK, B_N), 256, 0, stream>>>(poses, W, V);
    k_aw<<<(B_N * BKK * WW + 255) / 256, 256, 0, stream>>>(act, aw);

    // Phase 2: 3 fused EM iterations
    for (int it = 0; it < 3; ++it) {
        const int wb = it & 1;       // write buffer
        const int rb = wb ^ 1;       // read buffer (prev iteration)
        if (it > 0) {
            k_apsum<<<dim3(BKK, B_N), 256, 0, stream>>>(
                V, muB[rb], isB[rb], cB[rb], aB[rb], apsum);
        }
        k_mstep<<<dim3(CWW, B_N), 256, 0, stream>>>(
            V, aw, apsum,
            muB[rb], isB[rb], cB[rb], aB[rb],
            muB[wb], isB[wb], cB[wb], aB[wb],
            beta_v, beta_a, lambda_,
            out_poses, out_act, it);
    }
}